// GraphAttentionPooling_44220983280247
// MI455X (gfx1250) — compile-verified
//
#include <hip/hip_runtime.h>
#include <math.h>

typedef __attribute__((ext_vector_type(2))) float v2f;
typedef __attribute__((ext_vector_type(8))) float v8f;

#define DIM 256
#define ROWS_PER_BLOCK 256

// ---------------------------------------------------------------- init
__global__ __launch_bounds__(256) void zero_kernel(float* __restrict__ sums,
                                                   float* __restrict__ counts,
                                                   float* __restrict__ out,
                                                   int gd, int g) {
  int i = blockIdx.x * blockDim.x + threadIdx.x;
  if (i < gd) { sums[i] = 0.0f; out[i] = 0.0f; }
  if (i < g)  counts[i] = 0.0f;
}

// ------------------------------------------------- pass 1: segment sums
// batch is sorted: accumulate contiguous runs in registers, atomic-flush
// only at segment boundaries (~2 flushes per 256-row block).
__global__ __launch_bounds__(256) void segsum_kernel(const float* __restrict__ x,
                                                     const int* __restrict__ batch,
                                                     float* __restrict__ sums,
                                                     float* __restrict__ counts,
                                                     int N) {
  const int d = threadIdx.x;                 // one thread per column
  long start = (long)blockIdx.x * ROWS_PER_BLOCK;
  long end = start + ROWS_PER_BLOCK; if (end > N) end = N;
  if (start >= end) return;
  int curSeg = batch[start];
  float acc = 0.0f, cnt = 0.0f;
  for (long n = start; n < end; ++n) {
    int seg = batch[n];
    if (seg != curSeg) {
      atomicAdd(&sums[(long)curSeg * DIM + d], acc);
      if (d == 0) atomicAdd(&counts[curSeg], cnt);
      acc = 0.0f; cnt = 0.0f; curSeg = seg;
    }
    acc += x[n * DIM + d];
    cnt += 1.0f;
  }
  atomicAdd(&sums[(long)curSeg * DIM + d], acc);
  if (d == 0) atomicAdd(&counts[curSeg], cnt);
}

// ---------------------------------------------------------- mean (in place)
__global__ __launch_bounds__(256) void mean_kernel(float* __restrict__ sums,
                                                   const float* __restrict__ counts) {
  int g = blockIdx.x, d = threadIdx.x;
  float c = counts[g]; if (c < 1.0f) c = 1.0f;
  sums[(long)g * DIM + d] /= c;
}

// ---------------------------------------- GEMM + tanh via f32 WMMA (exact)
// One wave computes one 16x16 tile of transformed = tanh(mean @ W),
// accumulating 64 chained v_wmma_f32_16x16x4_f32 over K=256.
__global__ __launch_bounds__(32) void gemm_tanh_kernel(const float* __restrict__ mean,
                                                       const float* __restrict__ W,
                                                       float* __restrict__ transformed) {
  const int tileN = blockIdx.x;            // DIM/16 = 16 tiles
  const int tileM = blockIdx.y;            // G/16 tiles
  const int lane = threadIdx.x;
  const int half = lane >> 4;              // 0: K={0,1}, 1: K={2,3}
  const int l15  = lane & 15;
  const int rowA = tileM * 16 + l15;       // A: M striped across lanes
  const int colB = tileN * 16 + l15;       // B: N striped across lanes
  const int kOff = half * 2;
  v8f c = {};
  for (int k = 0; k < DIM; k += 4) {
    // A 16x4: v0 = K=kOff, v1 = K=kOff+1 (contiguous pair -> b64 load)
    v2f a = *(const v2f*)(mean + (long)rowA * DIM + k + kOff);
    // B 4x16: row K = k+kOff(+1), col N = colB
    v2f b;
    b.x = W[(long)(k + kOff)     * DIM + colB];
    b.y = W[(long)(k + kOff + 1) * DIM + colB];
    c = __builtin_amdgcn_wmma_f32_16x16x4_f32(false, a, false, b,
                                              (short)0, c, false, false);
  }
  // C/D layout: VGPR r -> row = r + 8*half, col = l15
#pragma unroll
  for (int r = 0; r < 8; ++r) {
    int row = tileM * 16 + half * 8 + r;
    transformed[(long)row * DIM + tileN * 16 + l15] = tanhf(c[r]);
  }
}

// ------------------------- pass 2: fused coef + weighted segment-sum
__global__ __launch_bounds__(256) void attn_kernel(const float* __restrict__ x,
                                                   const int* __restrict__ batch,
                                                   const float* __restrict__ transformed,
                                                   float* __restrict__ out,
                                                   int N) {
  __shared__ float red[8];
  const int d = threadIdx.x;
  const int wave = d >> 5, lane = d & 31;
  long start = (long)blockIdx.x * ROWS_PER_BLOCK;
  long end = start + ROWS_PER_BLOCK; if (end > N) end = N;
  if (start >= end) return;                // uniform per block
  int curSeg = batch[start];
  float acc = 0.0f;
  for (long n = start; n < end; ++n) {
    int seg = batch[n];
    if (seg != curSeg) {
      atomicAdd(&out[(long)curSeg * DIM + d], acc);
      acc = 0.0f; curSeg = seg;
    }
    float xv = x[n * DIM + d];
    float tv = transformed[(long)seg * DIM + d];   // 1MB, L2-hot
    float p = xv * tv;
#pragma unroll
    for (int off = 16; off > 0; off >>= 1)         // wave32 reduce
      p += __shfl_down(p, off, 32);
    if (lane == 0) red[wave] = p;
    __syncthreads();
    float s = red[0] + red[1] + red[2] + red[3] +
              red[4] + red[5] + red[6] + red[7];
    __syncthreads();
    float coef = 1.0f / (1.0f + expf(-10.0f * s));
    acc += coef * xv;
  }
  atomicAdd(&out[(long)curSeg * DIM + d], acc);
}

// ---------------------------------------------------------------- launch
extern "C" void kernel_launch(void* const* d_in, const int* in_sizes, int n_in,
                              void* d_out, int out_size, void* d_ws, size_t ws_size,
                              hipStream_t stream) {
  (void)n_in; (void)ws_size;
  const float* x     = (const float*)d_in[0];   // [N, 256]
  const float* W     = (const float*)d_in[1];   // [256, 256]
  const int*   batch = (const int*)d_in[2];     // [N], sorted
  const int N = in_sizes[0] / DIM;
  const int G = out_size / DIM;

  float* sums        = (float*)d_ws;                    // G*DIM (becomes mean)
  float* transformed = sums + (size_t)G * DIM;          // G*DIM
  float* counts      = transformed + (size_t)G * DIM;   // G
  float* out         = (float*)d_out;

  const int gd = G * DIM;
  zero_kernel<<<(gd + 255) / 256, 256, 0, stream>>>(sums, counts, out, gd, G);

  const int nblocks = (N + ROWS_PER_BLOCK - 1) / ROWS_PER_BLOCK;
  segsum_kernel<<<nblocks, 256, 0, stream>>>(x, batch, sums, counts, N);

  mean_kernel<<<G, 256, 0, stream>>>(sums, counts);

  dim3 ggrid(DIM / 16, (G + 15) / 16);
  gemm_tanh_kernel<<<ggrid, 32, 0, stream>>>(sums, W, transformed);

  attn_kernel<<<nblocks, 256, 0, stream>>>(x, batch, transformed, out, N);
}